// SessConv_13331578487504
// MI455X (gfx1250) — compile-verified
//
#include <hip/hip_runtime.h>
#include <hip/hip_bf16.h>

// ---------------------------------------------------------------------------
// SessConv on MI455X (gfx1250), wave32 + WMMA f32 path.
// x_{i+1} = D @ (A @ (x_i W_i^T));  acc += x_{i+1}/||x_{i+1}||;  out = acc/4
// All GEMMs via V_WMMA_F32_16X16X4_F32 (f32 in, f32 acc: exact ref precision).
// B operand staged in LDS as K-pair-interleaved layout so each lane's v2f
// operand is a single conflict-free ds_load_b64.
// ---------------------------------------------------------------------------

typedef __attribute__((ext_vector_type(2))) float v2f;
typedef __attribute__((ext_vector_type(8))) float v8f;

#define EMB      256
#define BATCH    4096
#define MAXLEN   50
#define LAYERS   3

#define BM 128           // block M tile
#define BN 32            // block N tile (two WMMA subtiles per wave)
#define BK 64            // K chunk staged in LDS
#define SA_LD (BK + 4)   // padded LDS row stride for A (floats)
#define SB_LD 96         // pair-row stride for B (floats): r even -> banks 2n,
                         // r odd -> banks 2n+32  => conflict-free b64 reads

// ---------------- gather + mean pool: seeds x and acc ----------------------
__global__ __launch_bounds__(256) void pool_kernel(
    const float* __restrict__ item_emb, const int* __restrict__ sess,
    const float* __restrict__ slen, float* __restrict__ x,
    float* __restrict__ acc) {
  const int b = blockIdx.x;
  const int e = threadIdx.x;  // EMB == 256 == blockDim.x
  const int* row = sess + (size_t)b * MAXLEN;
  float s = 0.0f;
  for (int l = 0; l < MAXLEN; ++l) {
    const int idx = row[l];                 // 0 is the zero pad row
    if (idx > 0) s += item_emb[(size_t)(idx - 1) * EMB + e];
  }
  s /= slen[b];
  x[(size_t)b * EMB + e] = s;
  acc[(size_t)b * EMB + e] = s;
}

// ---------------- tiled WMMA f32 GEMM --------------------------------------
// C[M,N] = A[M,K] @ B          (TRANSB=0: B is [K,N] row-major)
//                              (TRANSB=1: B is [N,K] row-major, i.e. A @ B^T)
// Block: 256 threads = 8 waves; each wave owns a 16x32 strip (two 16x16 WMMA
// accumulators sharing the A operand). Block covers 128(M) x 32(N).
template <bool TRANSB>
__global__ __launch_bounds__(256) void gemm_wmma(
    const float* __restrict__ A, const float* __restrict__ B,
    float* __restrict__ C, int M, int N, int K) {
  __shared__ float sA[BM * SA_LD];          // 34.0 KB
  __shared__ float sB[(BK / 2) * SB_LD];    // 12.0 KB, [kpair][n][2]

  const int tid  = threadIdx.x;
  const int lane = tid & 31;
  const int wave = tid >> 5;          // 0..7 -> M sub-tile
  const int m0   = blockIdx.y * BM;
  const int n0   = blockIdx.x * BN;
  const int ml   = lane & 15;         // A: row M%16 ; B/C: column N%16
  const int kh   = lane >> 4;         // K pair {0,1} vs {2,3}; C row +8

  v8f c0 = {};                        // cols n0 + ml
  v8f c1 = {};                        // cols n0 + 16 + ml

  for (int k0 = 0; k0 < K; k0 += BK) {
    // ---- stage A panel: BM x BK floats, float4 coalesced along K ----
    for (int j = 0; j < 8; ++j) {
      const int idx = tid + j * 256;   // 0..2047 float4 slots
      const int r   = idx >> 4;        // 16 float4 per row
      const int c4  = idx & 15;
      const float4 v =
          *(const float4*)(A + (size_t)(m0 + r) * K + k0 + c4 * 4);
      *(float4*)(sA + r * SA_LD + c4 * 4) = v;
    }
    // ---- stage B panel: BK x BN floats, pair-interleaved [kpair][n][2] ----
    if (!TRANSB) {
      // B is [K,N]: gather the (k, k+1) pair for one column per slot
      for (int j = 0; j < 4; ++j) {
        const int idx = tid + j * 256;  // 0..1023 pairs
        const int p   = idx >> 5;       // kpair 0..31
        const int n   = idx & 31;       // col 0..31
        const float v0 = B[(size_t)(k0 + 2 * p + 0) * N + n0 + n];
        const float v1 = B[(size_t)(k0 + 2 * p + 1) * N + n0 + n];
        *(v2f*)(sB + p * SB_LD + n * 2) = (v2f){v0, v1};
      }
    } else {
      // B is [N,K]: float4 along K, split into two contiguous pairs
      for (int j = 0; j < 2; ++j) {
        const int idx = tid + j * 256;  // 0..511 float4 slots
        const int nn  = idx >> 4;       // col 0..31
        const int kt  = (idx & 15) * 4; // k 0..60
        const float4 v =
            *(const float4*)(B + (size_t)(n0 + nn) * K + k0 + kt);
        *(v2f*)(sB + ((kt >> 1) + 0) * SB_LD + nn * 2) = (v2f){v.x, v.y};
        *(v2f*)(sB + ((kt >> 1) + 1) * SB_LD + nn * 2) = (v2f){v.z, v.w};
      }
    }
    // prefetch next A panel into the cache hierarchy while we compute
    if (k0 + BK < K)
      __builtin_prefetch(A + (size_t)(m0 + (tid >> 1)) * K + k0 + BK +
                             (tid & 1) * 32, 0, 3);
    __syncthreads();

    // ---- WMMA inner loop: 2 x (16x16x4) per K-step, shared A operand ----
    const float* pa = sA + (wave * 16 + ml) * SA_LD;
    for (int kk = 0; kk < BK; kk += 4) {
      const int p = (kk >> 1) + kh;    // this lane's kpair row
      // A 16x4: lanes 0-15 hold K={kk,kk+1}, lanes 16-31 K={kk+2,kk+3}
      const v2f a  = *(const v2f*)(pa + kk + 2 * kh);
      const v2f b0 = *(const v2f*)(sB + p * SB_LD + ml * 2);
      const v2f b1 = *(const v2f*)(sB + p * SB_LD + (16 + ml) * 2);
      c0 = __builtin_amdgcn_wmma_f32_16x16x4_f32(
          false, a, false, b0, (short)0, c0, false, false);
      c1 = __builtin_amdgcn_wmma_f32_16x16x4_f32(
          false, a, false, b1, (short)0, c1, false, false);
    }
    __syncthreads();
  }

  // ---- store C: VGPR r -> row (base + r) lanes 0-15, (base+8+r) lanes 16-31
  const int crow = m0 + wave * 16 + kh * 8;
  for (int r = 0; r < 8; ++r) {
    C[(size_t)(crow + r) * N + n0 + ml]      = c0[r];
    C[(size_t)(crow + r) * N + n0 + 16 + ml] = c1[r];
  }
}

// ---------------- row-normalize and accumulate -----------------------------
__global__ __launch_bounds__(256) void norm_acc_kernel(
    const float* __restrict__ x, float* __restrict__ acc) {
  __shared__ float red[256];
  const int b = blockIdx.x;
  const int e = threadIdx.x;
  const float v = x[(size_t)b * EMB + e];
  red[e] = v * v;
  __syncthreads();
  for (int s = 128; s > 0; s >>= 1) {
    if (e < s) red[e] += red[e + s];
    __syncthreads();
  }
  const float norm = fmaxf(sqrtf(red[0]), 1e-12f);
  acc[(size_t)b * EMB + e] += v / norm;
}

// ---------------- final scale ----------------------------------------------
__global__ __launch_bounds__(256) void scale_kernel(
    const float* __restrict__ acc, float* __restrict__ out) {
  const size_t i = (size_t)blockIdx.x * 256 + threadIdx.x;
  out[i] = acc[i] * (1.0f / (LAYERS + 1));
}

// ---------------------------------------------------------------------------
extern "C" void kernel_launch(void* const* d_in, const int* in_sizes, int n_in,
                              void* d_out, int out_size, void* d_ws,
                              size_t ws_size, hipStream_t stream) {
  const float* item_emb = (const float*)d_in[0];   // [50000, 256]
  const float* Dm       = (const float*)d_in[1];   // [4096, 4096]
  const float* Am       = (const float*)d_in[2];   // [4096, 4096]
  const int*   sess     = (const int*)  d_in[3];   // [4096, 50]
  const float* slen     = (const float*)d_in[4];   // [4096, 1]
  const float* wsess    = (const float*)d_in[5];   // [3, 256, 256]
  float* out = (float*)d_out;                      // [4096, 256]

  float* ws  = (float*)d_ws;                       // 4 x 4MB buffers
  const size_t XSZ = (size_t)BATCH * EMB;          // 1,048,576 floats
  float* x   = ws + 0 * XSZ;
  float* y   = ws + 1 * XSZ;
  float* t   = ws + 2 * XSZ;
  float* acc = ws + 3 * XSZ;

  pool_kernel<<<BATCH, 256, 0, stream>>>(item_emb, sess, slen, x, acc);

  const dim3 grid(EMB / BN, BATCH / BM);           // (8, 32)
  for (int i = 0; i < LAYERS; ++i) {
    // y = x @ W_i^T           [4096,256] x [256,256]^T
    gemm_wmma<true><<<grid, 256, 0, stream>>>(
        x, wsess + (size_t)i * EMB * EMB, y, BATCH, EMB, EMB);
    // t = A @ y               [4096,4096] x [4096,256]
    gemm_wmma<false><<<grid, 256, 0, stream>>>(Am, y, t, BATCH, EMB, BATCH);
    // x = D @ t               [4096,4096] x [4096,256]
    gemm_wmma<false><<<grid, 256, 0, stream>>>(Dm, t, x, BATCH, EMB, BATCH);
    // acc += x / max(||x||, eps)
    norm_acc_kernel<<<BATCH, 256, 0, stream>>>(x, acc);
  }
  scale_kernel<<<BATCH, 256, 0, stream>>>(acc, out);
}